// MLGANN_83554293776852
// MI455X (gfx1250) — compile-verified
//
#include <hip/hip_runtime.h>

// ---------------------------------------------------------------------------
// MLGANN (3-layer GCN + layer-attention pooling) for MI455X / gfx1250.
// fp32 WMMA (V_WMMA_F32_16X16X4_F32) for all dense GEMMs with dual
// accumulators (XDL ILP) and LDS-staged A strips; atomics + prefetch for the
// segment-sum edge aggregation (memory-bound part, ~800MB of L2 traffic).
// ---------------------------------------------------------------------------

typedef __attribute__((ext_vector_type(2))) float v2f;
typedef __attribute__((ext_vector_type(8))) float v8f;

#define N_NODES 50000
#define IN_DIM 256
#define HID 128
#define OUT_DIM 64
#define N_EDGES 800000
#define BATCH 4096
#define BN_EPS 1e-5f
#define LRELU 0.01f
#define LDS_PAD 4   // row stride K+4 floats -> bank stride 4 -> conflict-free

// ---------------------------------------------------------------------------
// C[M,N] = A[M,K] @ W[N,K]^T (+ optional bias[n]).
// grid.x = M/16 tiles, block = (32, N/16): one wave per 16x16 output tile.
// The 16xK A strip is staged once per block into padded LDS (all waves share
// it); K-loop uses two independent WMMA accumulator chains.
// ---------------------------------------------------------------------------
template <int K>
__global__ void gemm_wmma_f32(const float* __restrict__ A,
                              const float* __restrict__ W,
                              const float* __restrict__ bias,
                              float* __restrict__ C,
                              int N) {
  constexpr int KS = K + LDS_PAD;               // padded LDS row stride
  __shared__ float As[16 * KS];

  const int lane = threadIdx.x;                 // 0..31
  const int nt   = threadIdx.y;                 // n-tile
  const int mt   = blockIdx.x;                  // m-tile
  const int nl   = lane & 15;
  const int khalf = (lane >> 4) * 2;            // 0 or 2

  // cooperative stage of the 16xK A strip
  {
    const int nthreads = blockDim.x * blockDim.y;
    const int tid = threadIdx.y * 32 + threadIdx.x;
    const float* Ablk = A + (size_t)mt * 16 * K;
    for (int i = tid; i < 16 * K; i += nthreads) {
      int row = i / K;
      int col = i - row * K;
      As[row * KS + col] = Ablk[i];
    }
  }
  __syncthreads();

  const float* ap = &As[nl * KS + khalf];                    // LDS, b64 reads
  const float* bp = W + (size_t)(nt * 16 + nl) * K + khalf;  // B(k,n)=W[n][k]

  v8f acc0 = {0.f, 0.f, 0.f, 0.f, 0.f, 0.f, 0.f, 0.f};
  v8f acc1 = acc0;
#pragma unroll
  for (int k = 0; k < K; k += 8) {
    v2f a0 = *(const v2f*)(ap + k);
    v2f b0 = *(const v2f*)(bp + k);
    v2f a1 = *(const v2f*)(ap + k + 4);
    v2f b1 = *(const v2f*)(bp + k + 4);
    acc0 = __builtin_amdgcn_wmma_f32_16x16x4_f32(false, a0, false, b0,
                                                 (short)0, acc0, false, false);
    acc1 = __builtin_amdgcn_wmma_f32_16x16x4_f32(false, a1, false, b1,
                                                 (short)0, acc1, false, false);
  }
  v8f acc = acc0 + acc1;

  const int mhi = lane >> 4;                    // C/D: lanes>=16 hold M=r+8
#pragma unroll
  for (int r = 0; r < 8; ++r) {
    int m = mt * 16 + r + mhi * 8;
    int n = nt * 16 + nl;
    float v = acc[r];
    if (bias) v += bias[n];
    C[(size_t)m * N + n] = v;
  }
}

// ---------------------------------------------------------------------------
// Fused attention score: e[row] = sum_n q[n] * leaky_relu( (emb @ W^T)[row,n] )
// One wave per 16-row strip of emb_flat[3N,128]. The A fragments are
// preloaded into registers once (reused by all 8 n-tiles); dual WMMA chains.
// ---------------------------------------------------------------------------
__global__ void attn_scores_wmma(const float* __restrict__ emb,
                                 const float* __restrict__ W,
                                 const float* __restrict__ q,
                                 float* __restrict__ e) {
  const int lane  = threadIdx.x;
  const int mt    = blockIdx.x;
  const int nl    = lane & 15;
  const int khalf = (lane >> 4) * 2;

  const float* aptr = emb + (size_t)(mt * 16 + nl) * HID + khalf;
  v2f areg[HID / 4];
#pragma unroll
  for (int kk = 0; kk < HID / 4; ++kk)
    areg[kk] = *(const v2f*)(aptr + 4 * kk);

  float esum[8] = {0.f, 0.f, 0.f, 0.f, 0.f, 0.f, 0.f, 0.f};
#pragma unroll 2
  for (int ntile = 0; ntile < HID / 16; ++ntile) {
    const float* bptr = W + (size_t)(ntile * 16 + nl) * HID + khalf;
    v8f acc0 = {0.f, 0.f, 0.f, 0.f, 0.f, 0.f, 0.f, 0.f};
    v8f acc1 = acc0;
#pragma unroll
    for (int k = 0; k < HID; k += 8) {
      v2f b0 = *(const v2f*)(bptr + k);
      v2f b1 = *(const v2f*)(bptr + k + 4);
      acc0 = __builtin_amdgcn_wmma_f32_16x16x4_f32(false, areg[k / 4], false, b0,
                                                   (short)0, acc0, false, false);
      acc1 = __builtin_amdgcn_wmma_f32_16x16x4_f32(false, areg[k / 4 + 1], false, b1,
                                                   (short)0, acc1, false, false);
    }
    v8f acc = acc0 + acc1;
    float qn = q[ntile * 16 + nl];
#pragma unroll
    for (int r = 0; r < 8; ++r) {
      float hv = acc[r];
      hv = (hv > 0.f) ? hv : hv * LRELU;        // leaky_relu
      esum[r] += hv * qn;
    }
  }
  // reduce over the 16 column-lanes of each half-wave (bit 4 = M offset 8)
#pragma unroll
  for (int r = 0; r < 8; ++r) {
    float v = esum[r];
    v += __shfl_xor(v, 1, 32);
    v += __shfl_xor(v, 2, 32);
    v += __shfl_xor(v, 4, 32);
    v += __shfl_xor(v, 8, 32);
    if (nl == 0) e[mt * 16 + r + (lane >> 4) * 8] = v;
  }
}

// --------------------------- graph / elementwise ---------------------------
__global__ void deg_init_kernel(float* deg, int n) {
  int i = blockIdx.x * blockDim.x + threadIdx.x;
  if (i < n) deg[i] = 1.0f;                     // self-loop contribution
}

__global__ void deg_edge_kernel(const int* __restrict__ dst, float* deg, int ne) {
  int i = blockIdx.x * blockDim.x + threadIdx.x;
  if (i < ne) atomicAdd(&deg[dst[i]], 1.0f);
}

__global__ void dinv_kernel(float* deg, int n) {
  int i = blockIdx.x * blockDim.x + threadIdx.x;
  if (i < n) deg[i] = rsqrtf(deg[i]);
}

// agg[n,c] = h[n,c] * dinv[n]^2 + b[c]   (self-loop + bias, init for scatter)
__global__ void selfloop_bias_kernel(const float* __restrict__ h,
                                     const float* __restrict__ dinv,
                                     const float* __restrict__ b,
                                     float* __restrict__ agg, int total) {
  int i = blockIdx.x * blockDim.x + threadIdx.x;
  if (i >= total) return;
  int n = i >> 7;            // /HID
  int c = i & (HID - 1);
  float di = dinv[n];
  agg[i] = h[i] * di * di + b[c];
}

// segment_sum(h[src]*norm -> dst): 32 lanes per edge, 4 channels per lane.
// Prefetch the next edge's gather row (global_prefetch_b8) to hide the
// random-access latency behind the atomic stream.
__global__ void edge_agg_kernel(const float* __restrict__ h,
                                const int* __restrict__ src,
                                const int* __restrict__ dst,
                                const float* __restrict__ dinv,
                                float* __restrict__ agg, int ne) {
  long long t = (long long)blockIdx.x * blockDim.x + threadIdx.x;
  int e = (int)(t >> 5);
  if (e >= ne) return;
  int lane = (int)(t & 31);

  int enext = e + 8;                            // one block of edges ahead
  if (enext < ne) {
    int snext = src[enext];
    __builtin_prefetch(h + (size_t)snext * HID + lane * 4, 0, 3);
  }

  int s = src[e], d = dst[e];
  float norm = dinv[s] * dinv[d];
  const float* hs = h + (size_t)s * HID;
  float* ad = agg + (size_t)d * HID;
#pragma unroll
  for (int j = 0; j < HID / 32; ++j) {
    int c = lane + 32 * j;
    atomicAdd(&ad[c], hs[c] * norm);
  }
}

// per-channel batch stats of relu(v): stats[c]=mean, stats[128+c]=rsqrt(var+eps)
__global__ void bn_stats_kernel(const float* __restrict__ v,
                                float* __restrict__ stats, int n) {
  __shared__ float ss[256];
  __shared__ float ss2[256];
  int c = blockIdx.x;
  float s = 0.f, s2 = 0.f;
  for (int i = threadIdx.x; i < n; i += 256) {
    float x = fmaxf(v[(size_t)i * HID + c], 0.f);   // fused ReLU
    s += x;
    s2 += x * x;
  }
  ss[threadIdx.x] = s;
  ss2[threadIdx.x] = s2;
  __syncthreads();
  for (int stride = 128; stride > 0; stride >>= 1) {
    if (threadIdx.x < stride) {
      ss[threadIdx.x] += ss[threadIdx.x + stride];
      ss2[threadIdx.x] += ss2[threadIdx.x + stride];
    }
    __syncthreads();
  }
  if (threadIdx.x == 0) {
    float mean = ss[0] / (float)n;
    float var = ss2[0] / (float)n - mean * mean;    // biased variance
    stats[c] = mean;
    stats[HID + c] = rsqrtf(var + BN_EPS);
  }
}

// v = (relu(v)-mean)*inv*gamma + beta, in place
__global__ void bn_apply_kernel(float* __restrict__ v,
                                const float* __restrict__ stats,
                                const float* __restrict__ gamma,
                                const float* __restrict__ beta, int total) {
  int i = blockIdx.x * blockDim.x + threadIdx.x;
  if (i >= total) return;
  int c = i & (HID - 1);
  float x = fmaxf(v[i], 0.f);
  v[i] = (x - stats[c]) * stats[HID + c] * gamma[c] + beta[c];
}

// softmax over the 3 layers, in place (e -> alpha), layout [3,N]
__global__ void softmax3_kernel(float* __restrict__ e, int n) {
  int i = blockIdx.x * blockDim.x + threadIdx.x;
  if (i >= n) return;
  float a = e[i], b = e[n + i], c = e[2 * n + i];
  float m = fmaxf(a, fmaxf(b, c));
  float ea = expf(a - m), eb = expf(b - m), ec = expf(c - m);
  float inv = 1.f / (ea + eb + ec);
  e[i] = ea * inv;
  e[n + i] = eb * inv;
  e[2 * n + i] = ec * inv;
}

// z[n,h] = sum_l alpha[l,n] * emb[l,n,h]
__global__ void combine_kernel(const float* __restrict__ emb,
                               const float* __restrict__ alpha,
                               float* __restrict__ z, int n) {
  int i = blockIdx.x * blockDim.x + threadIdx.x;
  int total = n * HID;
  if (i >= total) return;
  int node = i >> 7;
  size_t stride = (size_t)n * HID;
  z[i] = alpha[node] * emb[i] +
         alpha[n + node] * emb[stride + i] +
         alpha[2 * n + node] * emb[2 * stride + i];
}

// out[b,c] = zout[ids[b], c]  (write-only result: non-temporal stores)
__global__ void gather_kernel(const float* __restrict__ zout,
                              const int* __restrict__ ids,
                              float* __restrict__ out, int batch) {
  int i = blockIdx.x * blockDim.x + threadIdx.x;
  if (i >= batch * OUT_DIM) return;
  int r = i >> 6;            // /OUT_DIM
  int c = i & (OUT_DIM - 1);
  float v = zout[(size_t)ids[r] * OUT_DIM + c];
  __builtin_nontemporal_store(v, &out[i]);
}

// ---------------------------------------------------------------------------
extern "C" void kernel_launch(void* const* d_in, const int* in_sizes, int n_in,
                              void* d_out, int out_size, void* d_ws, size_t ws_size,
                              hipStream_t stream) {
  (void)in_sizes; (void)n_in; (void)out_size; (void)ws_size;

  const float* x         = (const float*)d_in[0];
  const int*   edge      = (const int*)d_in[1];
  const int*   drug_pos  = (const int*)d_in[2];
  const int*   targ_pos  = (const int*)d_in[3];
  const int*   drug_neg  = (const int*)d_in[4];
  const int*   targ_neg  = (const int*)d_in[5];
  const float* gcn_w[3]  = {(const float*)d_in[7], (const float*)d_in[9],  (const float*)d_in[11]};
  const float* gcn_b[3]  = {(const float*)d_in[8], (const float*)d_in[10], (const float*)d_in[12]};
  const float* bn_gamma  = (const float*)d_in[13];
  const float* bn_beta   = (const float*)d_in[14];
  const float* W_D       = (const float*)d_in[15];
  const float* W_T       = (const float*)d_in[16];
  const float* q_D       = (const float*)d_in[17];
  const float* q_T       = (const float*)d_in[18];
  const float* out_w     = (const float*)d_in[19];
  const float* out_b     = (const float*)d_in[20];

  const int* src = edge;             // edge_index[0,:]
  const int* dst = edge + N_EDGES;   // edge_index[1,:]
  float* out = (float*)d_out;

  // -------- workspace carve-up (floats) --------
  float* ws    = (float*)d_ws;
  float* deg   = ws;                                   // 50048 (dinv after K3)
  float* stats = deg + 50048;                          // 256
  float* e     = stats + 256;                          // 3N = 150016 (alpha in place)
  float* h_tmp = e + 150016;                           // N*HID
  float* emb   = h_tmp + (size_t)N_NODES * HID;        // 3*N*HID
  float* zout  = emb + (size_t)3 * N_NODES * HID;      // N*OUT_DIM
  float* z     = h_tmp;                                // reuse after GCN layers

  const int MT = N_NODES / 16;        // 3125 exact
  const int MT3 = 3 * MT;             // 9375 strips of emb_flat
  const int NH = N_NODES * HID;

  // 1) degrees -> dinv
  deg_init_kernel<<<(N_NODES + 255) / 256, 256, 0, stream>>>(deg, N_NODES);
  deg_edge_kernel<<<(N_EDGES + 255) / 256, 256, 0, stream>>>(dst, deg, N_EDGES);
  dinv_kernel<<<(N_NODES + 255) / 256, 256, 0, stream>>>(deg, N_NODES);

  // 2) GCN layers
  for (int l = 0; l < 3; ++l) {
    float* embl = emb + (size_t)l * NH;

    if (l == 0) {
      gemm_wmma_f32<IN_DIM><<<dim3(MT), dim3(32, HID / 16), 0, stream>>>(
          x, gcn_w[0], nullptr, h_tmp, HID);
    } else {
      gemm_wmma_f32<HID><<<dim3(MT), dim3(32, HID / 16), 0, stream>>>(
          emb + (size_t)(l - 1) * NH, gcn_w[l], nullptr, h_tmp, HID);
    }
    selfloop_bias_kernel<<<(NH + 255) / 256, 256, 0, stream>>>(
        h_tmp, deg, gcn_b[l], embl, NH);
    edge_agg_kernel<<<((long long)N_EDGES * 32 + 255) / 256, 256, 0, stream>>>(
        h_tmp, src, dst, deg, embl, N_EDGES);
    bn_stats_kernel<<<HID, 256, 0, stream>>>(embl, stats, N_NODES);
    bn_apply_kernel<<<(NH + 255) / 256, 256, 0, stream>>>(
        embl, stats, bn_gamma, bn_beta, NH);
  }

  // 3) attention pooling + head: drug branch then target branch
  for (int branch = 0; branch < 2; ++branch) {
    const float* W = branch ? W_T : W_D;
    const float* q = branch ? q_T : q_D;

    attn_scores_wmma<<<MT3, 32, 0, stream>>>(emb, W, q, e);
    softmax3_kernel<<<(N_NODES + 255) / 256, 256, 0, stream>>>(e, N_NODES);
    combine_kernel<<<(NH + 255) / 256, 256, 0, stream>>>(emb, e, z, N_NODES);
    gemm_wmma_f32<HID><<<dim3(MT), dim3(32, OUT_DIM / 16), 0, stream>>>(
        z, out_w, out_b, zout, OUT_DIM);

    const int* ids0 = branch ? targ_pos : drug_pos;
    const int* ids1 = branch ? targ_neg : drug_neg;
    float* seg0 = out + (size_t)(branch ? 1 : 0) * BATCH * OUT_DIM;
    float* seg1 = out + (size_t)(branch ? 3 : 2) * BATCH * OUT_DIM;
    gather_kernel<<<(BATCH * OUT_DIM + 255) / 256, 256, 0, stream>>>(zout, ids0, seg0, BATCH);
    gather_kernel<<<(BATCH * OUT_DIM + 255) / 256, 256, 0, stream>>>(zout, ids1, seg1, BATCH);
  }
}